// InnerResShiftTriple_42451456753973
// MI455X (gfx1250) — compile-verified
//
#include <hip/hip_runtime.h>
#include <hip/hip_bf16.h>

typedef float v2f __attribute__((ext_vector_type(2)));
typedef float v8f __attribute__((ext_vector_type(8)));

#define NPIX 4096
#define BATCH 16

__device__ __forceinline__ v8f wmma_f32_4(v2f a, v2f b, v8f c) {
  // D = A(16x4 f32) * B(4x16 f32) + C(16x16 f32)
  return __builtin_amdgcn_wmma_f32_16x16x4_f32(false, a, false, b, (short)0, c, false, false);
}

// gfx1250 async Global->LDS DMA (ASYNCcnt-tracked), GV addressing mode.
__device__ __forceinline__ void async_g2l_b128(unsigned lds_off, const void* gaddr) {
  asm volatile("global_load_async_to_lds_b128 %0, %1, off"
               :: "v"(lds_off), "v"(gaddr) : "memory");
}
__device__ __forceinline__ void wait_async_le4() {   // allow 4 in flight (next tile)
  asm volatile("s_wait_asynccnt 0x4" ::: "memory");
}
__device__ __forceinline__ void wait_async_le1() {   // allow 1 in flight (next phase)
  asm volatile("s_wait_asynccnt 0x1" ::: "memory");
}
__device__ __forceinline__ unsigned lds_off32(const void* p) {
  return (unsigned)(size_t)p;   // low 32 bits of generic shared address = LDS byte offset
}

// ---------------- invnorm over channels of F_s ----------------
__global__ void rownorm_kernel(const float* __restrict__ x, float* __restrict__ invnorm) {
  int t = blockIdx.x * 256 + threadIdx.x;     // b*N + j
  int b = t >> 12, j = t & 4095;
  const float* p = x + ((size_t)(b * 512 + 256)) * NPIX + j;
  float s = 0.f;
  #pragma unroll 8
  for (int c = 0; c < 256; ++c) { float v = p[(size_t)c * NPIX]; s += v * v; }
  invnorm[t] = rsqrtf(s + 1e-8f);
}

// ---------------- lat_norm = F_s * invnorm (enables pure-copy async staging) ----------------
__global__ void latn_kernel(const float* __restrict__ x, const float* __restrict__ invnorm,
                            float* __restrict__ latn) {
  int t = blockIdx.x * 256 + threadIdx.x;     // over B*256*4096
  int p = t & 4095, c = (t >> 12) & 255, b = t >> 20;
  latn[t] = x[((size_t)(b * 512 + 256 + c)) * NPIX + p] * invnorm[b * NPIX + p];
}

// ---------------- fused similarity GEMM + masked argmax (double-buffered DMA) ----------------
__global__ void __launch_bounds__(256, 1)
argmax_kernel(const float* __restrict__ x, const float* __restrict__ latn,
              const int* __restrict__ flag, int* __restrict__ idx) {
  __shared__ float ldsB[2][256 * 16];         // 2 x 16 KB tiles, linear [K=c][N=j]
  int b = blockIdx.y;
  int tid = threadIdx.x, wave = tid >> 5, lane = tid & 31;
  int lj = lane & 15;
  int koff = (lane >> 4) << 1;                // 0 or 2
  int i0 = blockIdx.x * 128 + wave * 16;      // 16-row tile per wave
  const float* xb = x + (size_t)b * 512 * NPIX;
  unsigned ldsBase[2] = { lds_off32(&ldsB[0][0]), lds_off32(&ldsB[1][0]) };

  // A fragment: former rows i0..i0+15, full K=256, kept in registers (128 VGPRs)
  v2f a[64];
  #pragma unroll
  for (int k = 0; k < 64; ++k) {
    int c = 4 * k + koff;
    a[k].x = xb[(size_t)c * NPIX + i0 + lj];
    a[k].y = xb[(size_t)(c + 1) * NPIX + i0 + lj];
  }

  float best[8]; int bestj[8];
  #pragma unroll
  for (int v = 0; v < 8; ++v) { best[v] = -3.4e38f; bestj[v] = 0; }

  const float* latnb = latn + (size_t)b * 256 * NPIX;
  const int* flg = flag + b * NPIX;
  int stc = tid >> 2, stj = (tid & 3) << 2;   // staging decomposition: 16B per lane-op

  // prologue: DMA tile 0
  #pragma unroll
  for (int q = 0; q < 4; ++q) {
    int c = stc + (q << 6);
    async_g2l_b128(ldsBase[0] + (unsigned)((c * 16 + stj) << 2),
                   latnb + (size_t)c * NPIX + stj);
  }

  for (int jt = 0; jt < 256; ++jt) {
    __syncthreads();                          // everyone done reading buf[(jt-1)&1]
    {   // prefetch tile jt+1 (clamped; final dummy writes an unread buffer)
      int j0n = (jt < 255 ? jt + 1 : 255) * 16;
      unsigned bn = ldsBase[(jt + 1) & 1];
      #pragma unroll
      for (int q = 0; q < 4; ++q) {
        int c = stc + (q << 6);
        async_g2l_b128(bn + (unsigned)((c * 16 + stj) << 2),
                       latnb + (size_t)c * NPIX + j0n + stj);
      }
    }
    wait_async_le4();                         // tile jt's 4 DMAs retired (in-order)
    __syncthreads();
    const float* tb = &ldsB[jt & 1][0];
    v8f acc = {};
    #pragma unroll
    for (int k = 0; k < 64; ++k) {
      v2f bf;
      bf.x = tb[(4 * k + koff) * 16 + lj];
      bf.y = tb[(4 * k + koff + 1) * 16 + lj];
      acc = wmma_f32_4(a[k], bf, acc);
    }
    int j0 = jt * 16;
    bool masked = (flg[j0 + lj] == 1);
    int jcol = j0 + lj;
    #pragma unroll
    for (int v = 0; v < 8; ++v) {
      float val = masked ? -1.0e9f : acc[v];
      if (val > best[v]) { best[v] = val; bestj[v] = jcol; }
    }
  }
  // cross-lane argmax within each 16-lane half (ties -> smaller j, matches jnp.argmax)
  #pragma unroll
  for (int v = 0; v < 8; ++v) {
    #pragma unroll
    for (int off = 8; off >= 1; off >>= 1) {
      float ov = __shfl_xor(best[v], off, 16);
      int oj = __shfl_xor(bestj[v], off, 16);
      if (ov > best[v] || (ov == best[v] && oj < bestj[v])) { best[v] = ov; bestj[v] = oj; }
    }
  }
  if (lj == 0) {
    int mbase = i0 + ((lane >> 4) << 3);
    #pragma unroll
    for (int v = 0; v < 8; ++v) idx[b * NPIX + mbase + v] = bestj[v];
  }
}

// ---------------- F_fuse = F_c * gather(F_s, idx) gated by flag ----------------
__global__ void fuse_kernel(const float* __restrict__ x, const int* __restrict__ idx,
                            const int* __restrict__ flag, float* __restrict__ fuse) {
  int t = blockIdx.x * 256 + threadIdx.x;     // over B*256*4096
  int p = t & 4095, c = (t >> 12) & 255, b = t >> 20;
  float r = 0.f;
  if (flag[b * NPIX + p] == 1) {
    int j = idx[b * NPIX + p];
    r = x[((size_t)(b * 512 + c)) * NPIX + p] * x[((size_t)(b * 512 + 256 + c)) * NPIX + j];
  }
  fuse[t] = r;
}

// ---------------- weight transpose: w[cout][cin][r][s] -> wt[rs][cin][cout] ----------------
__global__ void transw_kernel(const float* __restrict__ w, float* __restrict__ wt, int Cin) {
  int t = blockIdx.x * 256 + threadIdx.x;
  int total = 256 * Cin * 9;
  if (t >= total) return;
  int rs = t % 9, cin = (t / 9) % Cin, cout = t / (9 * Cin);
  wt[((size_t)rs * Cin + cin) * 256 + cout] = w[t];
}

// ---------------- implicit-GEMM 3x3 conv + fused IN statistics (double-buffered) ----------------
__global__ void __launch_bounds__(256, 1)
conv_kernel(const float* __restrict__ inA, int bsA,
            const float* __restrict__ inB, int bsB, int split, int Cin,
            const float* __restrict__ wt, const float* __restrict__ bias,
            float* __restrict__ out, float* __restrict__ ssum, float* __restrict__ ssq) {
  __shared__ float ldsW[2][64 * 16];          // 2 x 4 KB weight stages, linear [cin][cout]
  __shared__ float lsum[16], lsq[16];
  int b = blockIdx.z;
  int c0 = blockIdx.y * 16;
  int tid = threadIdx.x, wave = tid >> 5, lane = tid & 31;
  int lj = lane & 15;
  int koff = (lane >> 4) << 1;
  int p0 = blockIdx.x * 128 + wave * 16;      // 16 pixels along x, one image row
  int py = p0 >> 6;
  int px = (p0 & 63) + lj;
  if (tid < 16) { lsum[tid] = 0.f; lsq[tid] = 0.f; }
  unsigned ldsBase[2] = { lds_off32(&ldsW[0][0]), lds_off32(&ldsW[1][0]) };
  const float* inAb = inA + (size_t)b * bsA;
  const float* inBb = inB + (size_t)b * bsB;
  v8f acc = {};
  int nchunk = Cin >> 6;
  int csh = (Cin == 512) ? 3 : 2;             // log2(nchunk)
  int nph = 9 << csh;
  int ci = tid >> 2, cq = (tid & 3) << 2;     // staging decomposition (1 B128 per thread)
  unsigned stoff = (unsigned)((ci * 16 + cq) << 2);

  // prologue: stage phase 0
  async_g2l_b128(ldsBase[0] + stoff, wt + (size_t)ci * 256 + c0 + cq);

  for (int ph = 0; ph < nph; ++ph) {
    int rs = ph >> csh, cbase = (ph & (nchunk - 1)) << 6;
    __syncthreads();
    {   // prefetch phase ph+1 (clamped)
      int phn = (ph < nph - 1) ? ph + 1 : ph;
      int rsn = phn >> csh, cbn = (phn & (nchunk - 1)) << 6;
      async_g2l_b128(ldsBase[(ph + 1) & 1] + stoff,
                     wt + (size_t)(rsn * Cin + cbn + ci) * 256 + c0 + cq);
    }
    wait_async_le1();
    __syncthreads();
    const float* tw = &ldsW[ph & 1][0];
    int dy = rs / 3 - 1, dx = rs % 3 - 1;
    int qy = py + dy, qx = px + dx;
    bool ok = (qy >= 0) && (qy < 64) && (qx >= 0) && (qx < 64);
    int qp = qy * 64 + qx;
    #pragma unroll
    for (int k = 0; k < 16; ++k) {
      v2f af;
      af.x = tw[(4 * k + koff) * 16 + lj];
      af.y = tw[(4 * k + koff + 1) * 16 + lj];
      int c = cbase + 4 * k + koff;
      v2f bf;
      if (ok) {
        const float* s0 = (c < split) ? (inAb + (size_t)c * NPIX)
                                      : (inBb + (size_t)(c - split) * NPIX);
        const float* s1 = ((c + 1) < split) ? (inAb + (size_t)(c + 1) * NPIX)
                                            : (inBb + (size_t)(c + 1 - split) * NPIX);
        bf.x = s0[qp]; bf.y = s1[qp];
      } else { bf.x = 0.f; bf.y = 0.f; }
      acc = wmma_f32_4(af, bf, acc);
    }
  }
  // epilogue: bias, store, per-channel sum/sumsq via 16-lane butterflies
  int mbase = (lane >> 4) << 3;               // 0 or 8
  #pragma unroll
  for (int v = 0; v < 8; ++v) {
    int c = c0 + mbase + v;
    float val = acc[v] + bias[c];
    out[((size_t)b * 256 + c) * NPIX + p0 + lj] = val;
    float s = val, q = val * val;
    #pragma unroll
    for (int off = 8; off >= 1; off >>= 1) {
      s += __shfl_xor(s, off, 16);
      q += __shfl_xor(q, off, 16);
    }
    if (lj == 0) { atomicAdd(&lsum[mbase + v], s); atomicAdd(&lsq[mbase + v], q); }
  }
  __syncthreads();
  if (tid < 16) {
    atomicAdd(&ssum[b * 256 + c0 + tid], lsum[tid]);
    atomicAdd(&ssq[b * 256 + c0 + tid], lsq[tid]);
  }
}

__global__ void finalize_kernel(const float* ssum, const float* ssq, float* mean, float* inv) {
  int t = blockIdx.x * 256 + threadIdx.x;
  if (t >= BATCH * 256) return;
  float m = ssum[t] * (1.0f / NPIX);
  float var = ssq[t] * (1.0f / NPIX) - m * m;
  mean[t] = m;
  inv[t] = rsqrtf(var + 1e-5f);
}

__global__ void normrelu_kernel(const float* __restrict__ y, const float* mean,
                                const float* inv, float* __restrict__ h) {
  size_t t = (size_t)blockIdx.x * 256 + threadIdx.x;
  int bc = (int)(t >> 12);
  float v = (y[t] - mean[bc]) * inv[bc];
  h[t] = v > 0.f ? v : 0.f;
}

__global__ void final_kernel(const float* __restrict__ x, const float* __restrict__ y,
                             const float* mean, const float* inv, float* __restrict__ outp) {
  size_t t = (size_t)blockIdx.x * 256 + threadIdx.x;   // over B*512*4096
  int p = (int)(t & 4095);
  int c = (int)((t >> 12) & 511);
  int b = (int)(t >> 21);
  float xv = x[t];
  float r = xv;
  if (c < 256) {
    int bc = b * 256 + c;
    r = xv + (y[(size_t)bc * NPIX + p] - mean[bc]) * inv[bc];
  }
  outp[t] = r;
}

__global__ void zero_kernel(float* p, int n) {
  int t = blockIdx.x * 256 + threadIdx.x;
  if (t < n) p[t] = 0.f;
}

extern "C" void kernel_launch(void* const* d_in, const int* in_sizes, int n_in,
                              void* d_out, int out_size, void* d_ws, size_t ws_size,
                              hipStream_t stream) {
  const float* x  = (const float*)d_in[0];
  const int* flag = (const int*)d_in[1];
  const float* w1 = (const float*)d_in[2];
  const float* b1 = (const float*)d_in[3];
  const float* w2 = (const float*)d_in[4];
  const float* b2 = (const float*)d_in[5];
  float* outp = (float*)d_out;

  char* ws = (char*)d_ws;
  size_t off = 0;
  auto alloc = [&](size_t bytes) { char* p = ws + off; off += (bytes + 255) & ~(size_t)255; return p; };
  float* bufA = (float*)alloc((size_t)BATCH * 256 * NPIX * 4);   // latn, then fuse, then h1
  float* bufY = (float*)alloc((size_t)BATCH * 256 * NPIX * 4);   // conv outputs
  float* wt1  = (float*)alloc((size_t)9 * 512 * 256 * 4);
  float* wt2  = (float*)alloc((size_t)9 * 256 * 256 * 4);
  float* invn = (float*)alloc((size_t)BATCH * NPIX * 4);
  int*   idx  = (int*)  alloc((size_t)BATCH * NPIX * 4);
  float* st   = (float*)alloc((size_t)8 * BATCH * 256 * 4);
  float *ssum1 = st,            *ssq1 = st + 4096,  *mean1 = st + 8192,  *inv1 = st + 12288;
  float *ssum2 = st + 16384,    *ssq2 = st + 20480, *mean2 = st + 24576, *inv2 = st + 28672;

  zero_kernel<<<dim3(128), 256, 0, stream>>>(st, 8 * BATCH * 256);
  transw_kernel<<<dim3(4608), 256, 0, stream>>>(w1, wt1, 512);
  transw_kernel<<<dim3(2304), 256, 0, stream>>>(w2, wt2, 256);
  rownorm_kernel<<<dim3(256), 256, 0, stream>>>(x, invn);
  latn_kernel<<<dim3(65536), 256, 0, stream>>>(x, invn, bufA);
  argmax_kernel<<<dim3(32, 16), 256, 0, stream>>>(x, bufA, flag, idx);
  fuse_kernel<<<dim3(65536), 256, 0, stream>>>(x, idx, flag, bufA);
  conv_kernel<<<dim3(32, 16, 16), 256, 0, stream>>>(x, 512 * NPIX, bufA, 256 * NPIX, 256, 512,
                                                    wt1, b1, bufY, ssum1, ssq1);
  finalize_kernel<<<dim3(16), 256, 0, stream>>>(ssum1, ssq1, mean1, inv1);
  normrelu_kernel<<<dim3(65536), 256, 0, stream>>>(bufY, mean1, inv1, bufA);
  conv_kernel<<<dim3(32, 16, 16), 256, 0, stream>>>(bufA, 256 * NPIX, bufA, 256 * NPIX, 256, 256,
                                                    wt2, b2, bufY, ssum2, ssq2);
  finalize_kernel<<<dim3(16), 256, 0, stream>>>(ssum2, ssq2, mean2, inv2);
  final_kernel<<<dim3(131072), 256, 0, stream>>>(x, bufY, mean2, inv2, outp);
}